// PAM_60198261621220
// MI455X (gfx1250) — compile-verified
//
#include <hip/hip_runtime.h>
#include <hip/hip_bf16.h>

typedef __attribute__((ext_vector_type(2))) float v2f;
typedef __attribute__((ext_vector_type(4))) float v4f;
typedef __attribute__((ext_vector_type(8))) float v8f;

// ---------------------------------------------------------------------------
// Prep 1: per-channel BN scale/shift  (scale = g/sqrt(v+eps), shift = b - m*scale)
// ---------------------------------------------------------------------------
__global__ void bn_scale_shift(const float* __restrict__ g, const float* __restrict__ bt,
                               const float* __restrict__ m, const float* __restrict__ v,
                               float* __restrict__ scale, float* __restrict__ shift) {
    int c = blockIdx.x * blockDim.x + threadIdx.x;
    if (c < 256) {
        float s = g[c] * rsqrtf(v[c] + 1e-5f);
        scale[c] = s;
        shift[c] = bt[c] - m[c] * s;
    }
}

// ---------------------------------------------------------------------------
// Prep 2: fold BN into conv1:  conv(BN(x)) = conv_{w*s}(x) + bias_fold
// ---------------------------------------------------------------------------
__global__ void fold_conv1(const float* __restrict__ w1, const float* __restrict__ b1,
                           const float* __restrict__ g, const float* __restrict__ bt,
                           const float* __restrict__ m, const float* __restrict__ v,
                           float* __restrict__ w1f, float* __restrict__ b1f) {
    int o = blockIdx.x;           // 0..255
    int t = threadIdx.x;          // 0..63 : input channel within group
    int grp = o >> 6;
    int ci = grp * 64 + t;
    float s  = g[ci] * rsqrtf(v[ci] + 1e-5f);
    float sh = bt[ci] - m[ci] * s;
    float acc = 0.f;
    for (int tap = 0; tap < 9; ++tap) {
        float wv = w1[(size_t)(o * 64 + t) * 9 + tap];
        w1f[(size_t)o * 576 + t * 9 + tap] = wv * s;
        acc += wv * sh;
    }
    __shared__ float red[64];
    red[t] = acc;
    __syncthreads();
    if (t == 0) {
        float sum = 0.f;
        for (int i = 0; i < 64; ++i) sum += red[i];
        b1f[o] = b1[o] + sum;
    }
}

// ---------------------------------------------------------------------------
// Prep 3: NCHW (B,64,H,W) -> NHWC (B,H,W,64) via LDS tile (for x_left/x_right)
// ---------------------------------------------------------------------------
__global__ __launch_bounds__(256) void nchw_to_nhwc(const float* __restrict__ in,
                                                    float* __restrict__ out) {
    constexpr int H = 64, W = 96, C = 64;
    __shared__ float t[C * W];
    int bh = blockIdx.x;
    int h = bh % H, b = bh / H;
    const float* ip = in + ((size_t)b * C * H + h) * W;
    for (int i = threadIdx.x; i < C * W; i += 256) {
        int c = i / W, w = i % W;
        t[c * W + w] = ip[(size_t)c * H * W + w];   // coalesced along w
    }
    __syncthreads();
    float* op = out + ((size_t)(b * H + h)) * W * C;
    for (int i = threadIdx.x; i < C * W; i += 256) {
        int w = i / C, c = i % C;
        op[i] = t[c * W + w];                       // coalesced along c
    }
}

// ---------------------------------------------------------------------------
// Grouped 3x3 conv (groups=4, 256->256) as implicit GEMM with f32 WMMA 16x16x4.
// MODE 0: conv1 (folded-BN weights) + LeakyReLU(0.1)
// MODE 1: conv2 + residual BN(catfea)
// ---------------------------------------------------------------------------
template <int MODE>
__global__ __launch_bounds__(256) void conv3x3_wmma(
    const float* __restrict__ in, const float* __restrict__ wgt,
    const float* __restrict__ bias, const float* __restrict__ cat,
    const float* __restrict__ scale, const float* __restrict__ shift,
    float* __restrict__ out) {
    constexpr int H = 64, W = 96, C4 = 256;
    __shared__ float lin[64 * 3 * 34];   // [ic][dr][cc] cc = col - col0 + 1

    int bx = blockIdx.x;
    int nt = bx % 3;  bx /= 3;
    int h  = bx % H;  bx /= H;
    int g  = bx % 4;  bx /= 4;
    int b  = bx;
    int col0 = nt * 32;
    int tid = threadIdx.x;

    const float* inB = in + ((size_t)(b * C4 + g * 64)) * H * W;
    for (int idx = tid; idx < 64 * 102; idx += 256) {
        int ic = idx / 102, rem = idx % 102, dr = rem / 34, cc = rem % 34;
        int r = h + dr - 1, c = col0 + cc - 1;
        float vv = 0.f;
        if (r >= 0 && r < H && c >= 0 && c < W) vv = inB[(size_t)ic * H * W + r * W + c];
        lin[idx] = vv;
    }
    __syncthreads();

    int wave = tid >> 5, lane = tid & 31;
    int mt  = wave & 3;          // out-channel tile within group
    int nt2 = wave >> 2;         // pixel half-tile (0/1)
    int l16 = lane & 15, kh = lane >> 4;
    int nn  = nt2 * 16 + l16;    // pixel within the 32-wide block tile
    const float* wrow = wgt + (size_t)(g * 64 + mt * 16 + l16) * 576;  // A row (m = lane%16)

    v8f acc = {};
    for (int kb = 0; kb < 576; kb += 4) {
        int kk = kb + kh * 2;                      // even -> 8B aligned
        v2f a = *(const v2f*)(wrow + kk);          // A[m, kk..kk+1]
        int ic0 = kk / 9,       t0 = kk - ic0 * 9;
        int ic1 = (kk + 1) / 9, t1 = (kk + 1) - ic1 * 9;
        v2f bv;
        bv.x = lin[ic0 * 102 + (t0 / 3) * 34 + nn + (t0 % 3)];
        bv.y = lin[ic1 * 102 + (t1 / 3) * 34 + nn + (t1 % 3)];
        acc = __builtin_amdgcn_wmma_f32_16x16x4_f32(false, a, false, bv, (short)0, acc,
                                                    false, false);
    }

    int col = col0 + nn;
    for (int r = 0; r < 8; ++r) {
        int oc = g * 64 + mt * 16 + kh * 8 + r;    // D: M = (lane/16)*8 + r
        size_t oi = ((size_t)(b * C4 + oc)) * H * W + h * W + col;
        float vv = acc[r] + bias[oc];
        if constexpr (MODE == 0) {
            vv = vv > 0.f ? vv : 0.1f * vv;        // LeakyReLU(0.1)
        } else {
            vv += cat[oi] * scale[oc] + shift[oc]; // + BN(catfea) residual
        }
        out[oi] = vv;
    }
}

// ---------------------------------------------------------------------------
// Grouped 1x1 conv (256 -> 64, groups=4): Q from res_left, K from res_right.
// 8 waves = 4 groups x {Q,K}; each wave M=16,K=64,N=16 with f32 WMMA.
// Output written in NHWC (B,H,W,64) via an LDS re-tile, fully coalesced.
// ---------------------------------------------------------------------------
__global__ __launch_bounds__(256) void qk1x1_wmma(
    const float* __restrict__ resL, const float* __restrict__ resR,
    const float* __restrict__ bqw, const float* __restrict__ bqb,
    const float* __restrict__ bsw, const float* __restrict__ bsb,
    float* __restrict__ Qb, float* __restrict__ Kb) {
    constexpr int H = 64, W = 96;
    __shared__ float smT[2 * 16 * 64];             // [which][pixel][oc]
    int bx = blockIdx.x;
    int nt = bx % 6;  bx /= 6;
    int h  = bx % H;  bx /= H;
    int b  = bx;
    int col0 = nt * 16;
    int tid = threadIdx.x, wave = tid >> 5, lane = tid & 31;
    int g = wave & 3, which = wave >> 2;

    const float* res = which ? resR : resL;
    const float* Wt  = which ? bsw : bqw;
    const float* Bi  = which ? bsb : bqb;

    int l16 = lane & 15, kh = lane >> 4;
    const float* wrow  = Wt + (size_t)(g * 16 + l16) * 64;
    const float* rbase = res + ((size_t)(b * 256 + g * 64)) * H * W + h * W + col0 + l16;

    v8f acc = {};
    for (int kb = 0; kb < 64; kb += 4) {
        int kk = kb + kh * 2;
        v2f a = *(const v2f*)(wrow + kk);
        v2f bv;
        bv.x = rbase[(size_t)kk * H * W];
        bv.y = rbase[(size_t)(kk + 1) * H * W];
        acc = __builtin_amdgcn_wmma_f32_16x16x4_f32(false, a, false, bv, (short)0, acc,
                                                    false, false);
    }
    for (int r = 0; r < 8; ++r) {
        int oc = g * 16 + kh * 8 + r;
        smT[which * 1024 + l16 * 64 + oc] = acc[r] + Bi[oc];
    }
    __syncthreads();
    size_t obase = ((size_t)(b * H + h) * W + col0) * 64;
    *(v4f*)(Qb + obase + tid * 4) = *(const v4f*)(smT + tid * 4);
    *(v4f*)(Kb + obase + tid * 4) = *(const v4f*)(smT + 1024 + tid * 4);
}

// ---------------------------------------------------------------------------
// Attention, wave per pixel. All operands NHWC so the channel axis is
// contiguous: score phase uses b128 loads per window position; value phase
// reads are coalesced across lanes. Mean-centering cancels in softmax.
// Window-validity quirks of the reference reproduced (h>=H-4 or center>=W-4
// -> zero patch -> uniform softmax; value gathers use r2l coords).
// ---------------------------------------------------------------------------
__device__ __forceinline__ float wred_max(float v) {
    for (int m = 16; m > 0; m >>= 1) v = fmaxf(v, __shfl_xor(v, m, 32));
    return v;
}
__device__ __forceinline__ float wred_sum(float v) {
    for (int m = 16; m > 0; m >>= 1) v += __shfl_xor(v, m, 32);
    return v;
}
__device__ __forceinline__ float dot4(v4f a, v4f b) {
    return fmaf(a.x, b.x, fmaf(a.y, b.y, fmaf(a.z, b.z, a.w * b.w)));
}

__global__ __launch_bounds__(256) void pam_attn(
    const float* __restrict__ Qb, const float* __restrict__ Kb,    // NHWC
    const float* __restrict__ xlT, const float* __restrict__ xrT,  // NHWC
    const int* __restrict__ dl_, const int* __restrict__ dr_,
    float* __restrict__ outL, float* __restrict__ outR) {          // NCHW
    constexpr int H = 64, W = 96, C = 64, HW = H * W;
    __shared__ float sm[8 * 256];
    int tid = threadIdx.x, wave = tid >> 5, lane = tid & 31;
    float* smQ  = sm + wave * 256;
    float* smK  = smQ + 64;
    float* smMR = smQ + 128;
    float* smML = smQ + 192;

    int pix = blockIdx.x * 8 + wave;
    int w = pix % W, h = (pix / W) % H, b = pix / (W * H);
    int dl = dl_[(b * H + h) * W + w];
    int dr = dr_[(b * H + h) * W + w];
    int c0r = max(w - dl, 0);
    int c0l = min(w + dr, W - 1);
    bool winR = (h < H - 4) && (c0r < W - 4);
    bool winL = (h < H - 4) && (c0l < W - 4);

    size_t pb = ((size_t)(b * H + h) * W + w) * C;       // NHWC pixel base
    smQ[lane]      = Qb[pb + lane];
    smQ[lane + 32] = Qb[pb + lane + 32];
    smK[lane]      = Kb[pb + lane];
    smK[lane + 32] = Kb[pb + lane + 32];
    __syncthreads();

    // window positions k0 = lane, k1 = lane+32
    int k0 = lane, k1 = lane + 32;
    int rk0 = h - 4 + (k0 >> 3), rk1 = h - 4 + (k1 >> 3);
    int cR0 = c0r - 4 + (k0 & 7), cR1 = c0r - 4 + (k1 & 7);
    int cL0 = c0l - 4 + (k0 & 7), cL1 = c0l - 4 + (k1 & 7);
    bool iR0 = winR && rk0 >= 0 && rk0 < H && cR0 >= 0 && cR0 < W;
    bool iR1 = winR && rk1 >= 0 && rk1 < H && cR1 >= 0 && cR1 < W;
    bool iL0 = winL && rk0 >= 0 && rk0 < H && cL0 >= 0 && cL0 < W;
    bool iL1 = winL && rk1 >= 0 && rk1 < H && cL1 >= 0 && cL1 < W;
    size_t base = (size_t)b * HW * C;
    size_t oR0 = iR0 ? base + ((size_t)rk0 * W + cR0) * C : 0;
    size_t oR1 = iR1 ? base + ((size_t)rk1 * W + cR1) * C : 0;
    size_t oL0 = iL0 ? base + ((size_t)rk0 * W + cL0) * C : 0;
    size_t oL1 = iL1 ? base + ((size_t)rk1 * W + cL1) * C : 0;

    float sR0 = 0, sR1 = 0, sL0 = 0, sL1 = 0;
    for (int c = 0; c < C; c += 4) {
        v4f q  = *(const v4f*)(smQ + c);
        v4f kc = *(const v4f*)(smK + c);
        v4f a0 = *(const v4f*)(Kb + oR0 + c);
        v4f a1 = *(const v4f*)(Kb + oR1 + c);
        v4f b0 = *(const v4f*)(Qb + oL0 + c);
        v4f b1 = *(const v4f*)(Qb + oL1 + c);
        sR0 += dot4(q, a0);
        sR1 += dot4(q, a1);
        sL0 += dot4(kc, b0);
        sL1 += dot4(kc, b1);
    }
    // zero out-of-window contributions once (reads above hit valid memory)
    sR0 = iR0 ? sR0 : 0.f;  sR1 = iR1 ? sR1 : 0.f;
    sL0 = iL0 ? sL0 : 0.f;  sL1 = iL1 ? sL1 : 0.f;

    // softmax across 64 positions (2 per lane)
    float mR = wred_max(fmaxf(sR0, sR1));
    float e0 = __expf(sR0 - mR), e1 = __expf(sR1 - mR);
    float sumR = wred_sum(e0 + e1);
    float mL = wred_max(fmaxf(sL0, sL1));
    float f0 = __expf(sL0 - mL), f1 = __expf(sL1 - mL);
    float sumL = wred_sum(f0 + f1);
    smMR[k0] = e0 / sumR;  smMR[k1] = e1 / sumR;
    smML[k0] = f0 / sumL;  smML[k1] = f1 / sumL;
    __syncthreads();

    // value gathers: BOTH use winR coords (reference NOTE); lane = channel pair
    float aL0 = 0, aL1 = 0, aR0 = 0, aR1 = 0;
    int c0 = lane, c1 = lane + 32;
    for (int k = 0; k < 64; ++k) {
        int rk = h - 4 + (k >> 3), ck = c0r - 4 + (k & 7);
        bool in = winR && rk >= 0 && rk < H && ck >= 0 && ck < W;
        size_t off = in ? base + ((size_t)rk * W + ck) * C : 0;
        float xr0 = xrT[off + c0];  xr0 = in ? xr0 : 0.f;   // coalesced across lanes
        float xr1 = xrT[off + c1];  xr1 = in ? xr1 : 0.f;
        float xl0 = xlT[off + c0];  xl0 = in ? xl0 : 0.f;
        float xl1 = xlT[off + c1];  xl1 = in ? xl1 : 0.f;
        float mr = smMR[k], ml = smML[k];
        aL0 = fmaf(mr, xr0, aL0);  aL1 = fmaf(mr, xr1, aL1);
        aR0 = fmaf(ml, xl0, aR0);  aR1 = fmaf(ml, xl1, aR1);
    }
    float VL = (w - dl >= 0) ? 1.f : 0.f;
    float VR = (w + dr <= W - 1) ? 1.f : 0.f;
    float xlc0 = xlT[pb + c0], xlc1 = xlT[pb + c1];
    float xrc0 = xrT[pb + c0], xrc1 = xrT[pb + c1];
    size_t o0 = ((size_t)(b * C + c0)) * HW + (size_t)h * W + w;   // NCHW out
    size_t o1 = ((size_t)(b * C + c1)) * HW + (size_t)h * W + w;
    outL[o0] = xlc0 + VL * aL0;
    outL[o1] = xlc1 + VL * aL1;
    outR[o0] = xrc0 + VR * aR0;
    outR[o1] = xrc1 + VR * aR1;
}

// ---------------------------------------------------------------------------
extern "C" void kernel_launch(void* const* d_in, const int* in_sizes, int n_in,
                              void* d_out, int out_size, void* d_ws, size_t ws_size,
                              hipStream_t stream) {
    const float* x_left   = (const float*)d_in[0];
    const float* x_right  = (const float*)d_in[1];
    const float* cat_l    = (const float*)d_in[2];
    const float* cat_r    = (const float*)d_in[3];
    const int*   d_left   = (const int*)d_in[4];
    const int*   d_right  = (const int*)d_in[5];
    const float* bn_gamma = (const float*)d_in[6];
    const float* bn_beta  = (const float*)d_in[7];
    const float* bn_mean  = (const float*)d_in[8];
    const float* bn_var   = (const float*)d_in[9];
    const float* rb_w1    = (const float*)d_in[10];
    const float* rb_b1    = (const float*)d_in[11];
    const float* rb_w2    = (const float*)d_in[12];
    const float* rb_b2    = (const float*)d_in[13];
    const float* bq_w     = (const float*)d_in[14];
    const float* bq_b     = (const float*)d_in[15];
    const float* bs_w     = (const float*)d_in[16];
    const float* bs_b     = (const float*)d_in[17];

    const int H = 64, W = 96;
    int BN = in_sizes[0] / (64 * H * W);         // batch (=2)

    float* ws    = (float*)d_ws;
    float* scale = ws;                 // 256
    float* shift = ws + 256;           // 256
    float* w1f   = ws + 512;           // 256*576 = 147456
    float* b1f   = ws + 512 + 147456;  // 256
    size_t NBIG  = (size_t)BN * 256 * H * W;
    size_t NQ    = (size_t)BN * 64 * H * W;
    float* S0  = ws + 148224;          // act1_L, later res_R
    float* S1  = S0 + NBIG;            // act1_R
    float* S2  = S1 + NBIG;            // res_L
    float* Qb  = S2 + NBIG;            // Q  (NHWC)
    float* Kb  = Qb + NQ;              // K  (NHWC)
    float* xlT = Kb + NQ;              // x_left  (NHWC)
    float* xrT = xlT + NQ;             // x_right (NHWC)

    bn_scale_shift<<<1, 256, 0, stream>>>(bn_gamma, bn_beta, bn_mean, bn_var, scale, shift);
    fold_conv1<<<256, 64, 0, stream>>>(rb_w1, rb_b1, bn_gamma, bn_beta, bn_mean, bn_var,
                                       w1f, b1f);
    nchw_to_nhwc<<<BN * H, 256, 0, stream>>>(x_left, xlT);
    nchw_to_nhwc<<<BN * H, 256, 0, stream>>>(x_right, xrT);

    dim3 cg(BN * 4 * H * 3);
    conv3x3_wmma<0><<<cg, 256, 0, stream>>>(cat_l, w1f, b1f, nullptr, nullptr, nullptr, S0);
    conv3x3_wmma<0><<<cg, 256, 0, stream>>>(cat_r, w1f, b1f, nullptr, nullptr, nullptr, S1);
    conv3x3_wmma<1><<<cg, 256, 0, stream>>>(S0, rb_w2, rb_b2, cat_l, scale, shift, S2);
    conv3x3_wmma<1><<<cg, 256, 0, stream>>>(S1, rb_w2, rb_b2, cat_r, scale, shift, S0);

    qk1x1_wmma<<<BN * H * 6, 256, 0, stream>>>(S2, S0, bq_w, bq_b, bs_w, bs_b, Qb, Kb);

    float* outL = (float*)d_out;
    float* outR = outL + NQ;
    pam_attn<<<BN * H * W / 8, 256, 0, stream>>>(Qb, Kb, xlT, xrT, d_left, d_right,
                                                 outL, outR);
}